// ImageGAEEncoder_67156108640280
// MI455X (gfx1250) — compile-verified
//
#include <hip/hip_runtime.h>
#include <hip/hip_bf16.h>
#include <math.h>

// ---------------------------------------------------------------------------
// GCN encoder for MI455X (gfx1250): fp32 WMMA 16x16x4 GEMMs + L2-resident
// edge scatter with fp32 global atomics. Bandwidth-bound; fp32 kept throughout.
// ---------------------------------------------------------------------------

typedef __attribute__((ext_vector_type(2))) float v2f;
typedef __attribute__((ext_vector_type(8))) float v8f;

#define IN_C 256
#define C1   128
#define C2   64
#define BN_EPS 1e-5f
#define NB_STATS 250   // stat-reduction blocks (250 * 200 rows = 50000)

// ---------------- degree / normalization ----------------

__global__ void fill_ones(float* __restrict__ p, int n) {
    int i = blockIdx.x * blockDim.x + threadIdx.x;
    if (i < n) p[i] = 1.0f;   // self-loop weight
}

__global__ void deg_accum(const int* __restrict__ col, const float* __restrict__ ew,
                          float* __restrict__ deg, int e) {
    int i = blockIdx.x * blockDim.x + threadIdx.x;
    if (i < e) atomicAdd(&deg[col[i]], ew[i]);
}

__global__ void rsqrt_ip(float* __restrict__ p, int n) {
    int i = blockIdx.x * blockDim.x + threadIdx.x;
    if (i < n) {
        float d = p[i];
        p[i] = (d > 0.0f) ? rsqrtf(d) : 0.0f;
    }
}

// ---------------- fp32 WMMA GEMM: C[M x COUT] = A[M x K] @ W[K x COUT] (+bias) ----
// Block = 128 threads = 4 waves; each wave owns a 16-row strip and loops the
// 16-wide N tiles. Fragment layout per CDNA5 ISA 7.12.2 (f32, 16x16x4):
//   A: lane(half,lm) holds A[lm][k0+2*half], A[lm][k0+2*half+1]
//   B: lane(half,lm) holds B[k0+2*half][lm], B[k0+2*half+1][lm]
//   C/D: acc[v] = D[v + 8*half][lm]
template<int K, int COUT>
__global__ __launch_bounds__(128) void gemm_wmma_f32(
        const float* __restrict__ A, const float* __restrict__ W,
        const float* __restrict__ bias, float* __restrict__ C, int M)
{
    const int wave = threadIdx.x >> 5;
    const int lane = threadIdx.x & 31;
    const int half = lane >> 4;
    const int lm   = lane & 15;
    const int m0   = blockIdx.x * 64 + wave * 16;
    if (m0 >= M) return;                       // wave-uniform: EXEC stays all-ones

    const float* __restrict__ Arow = A + (size_t)(m0 + lm) * K;

    for (int nt = 0; nt < COUT / 16; ++nt) {
        const int n0 = nt * 16;
        v8f acc = {};
        #pragma unroll 8
        for (int k = 0; k < K; k += 4) {
            v2f af, bf;
            af.x = Arow[k + 2 * half + 0];
            af.y = Arow[k + 2 * half + 1];
            const float* __restrict__ Wp = W + (size_t)(k + 2 * half) * COUT + n0 + lm;
            bf.x = Wp[0];
            bf.y = Wp[COUT];
            acc = __builtin_amdgcn_wmma_f32_16x16x4_f32(
                      false, af, false, bf, (short)0, acc, false, false);
        }
        const float bv = bias ? bias[n0 + lm] : 0.0f;
        #pragma unroll
        for (int v = 0; v < 8; ++v) {
            const int m = m0 + v + 8 * half;
            C[(size_t)m * COUT + n0 + lm] = acc[v] + bv;
        }
    }
}

// ---------------- GCN aggregation ----------------

// out[i,c] = h[i,c] * dinv[i]^2 + bias[c]   (self-loop term; also initializes out)
template<int C>
__global__ void selfloop_init(const float* __restrict__ h, const float* __restrict__ dinv,
                              const float* __restrict__ bias, float* __restrict__ out, int n) {
    int gid = blockIdx.x * blockDim.x + threadIdx.x;
    if (gid >= n * C) return;
    int row = gid / C, c = gid % C;
    float di = dinv[row];
    out[gid] = h[gid] * di * di + bias[c];
}

// out[col] += h[row] * dinv[row]*ew*dinv[col] ; one thread per (edge, 4 channels)
template<int C>
__global__ void edge_scatter(const float* __restrict__ h, const int* __restrict__ row,
                             const int* __restrict__ col, const float* __restrict__ ew,
                             const float* __restrict__ dinv, float* __restrict__ out, int e) {
    const int QP = C / 4;
    int gid = blockIdx.x * blockDim.x + threadIdx.x;
    if (gid >= e * QP) return;
    int eid = gid / QP;
    int q   = (gid % QP) * 4;
    int r  = row[eid];
    int cn = col[eid];
    float nrm = dinv[r] * ew[eid] * dinv[cn];
    const float4 hv = *reinterpret_cast<const float4*>(h + (size_t)r * C + q);
    float* dst = out + (size_t)cn * C + q;
    atomicAdd(dst + 0, hv.x * nrm);
    atomicAdd(dst + 1, hv.y * nrm);
    atomicAdd(dst + 2, hv.z * nrm);
    atomicAdd(dst + 3, hv.w * nrm);
}

// ---------------- BatchNorm (deterministic two-level reduction) ----------------

template<int C>
__global__ __launch_bounds__(256) void bn_partial(
        const float* __restrict__ a, int nRows, int rowsPerBlock,
        float* __restrict__ psum, float* __restrict__ psumsq) {
    const int T  = 256;
    const int RP = T / C;                 // rows handled in parallel per block
    int c = threadIdx.x % C;
    int r = threadIdx.x / C;
    int row0   = blockIdx.x * rowsPerBlock;
    int rowEnd = min(nRows, row0 + rowsPerBlock);
    float s = 0.0f, ss = 0.0f;
    for (int i = row0 + r; i < rowEnd; i += RP) {   // coalesced: lane c reads channel c
        float v = a[(size_t)i * C + c];
        s += v; ss += v * v;
    }
    __shared__ float ls[T];
    __shared__ float lss[T];
    ls[threadIdx.x] = s; lss[threadIdx.x] = ss;
    __syncthreads();
    if (r == 0) {                                    // fixed-order combine => deterministic
        for (int j = 1; j < RP; ++j) { s += ls[j * C + c]; ss += lss[j * C + c]; }
        psum  [(size_t)blockIdx.x * C + c] = s;
        psumsq[(size_t)blockIdx.x * C + c] = ss;
    }
}

template<int C>
__global__ void bn_finalize(const float* __restrict__ psum, const float* __restrict__ psumsq,
                            const float* __restrict__ gamma, const float* __restrict__ beta,
                            float* __restrict__ scale, float* __restrict__ shift,
                            int nRows, int nb) {
    int c = threadIdx.x;
    if (c >= C) return;
    float s = 0.0f, ss = 0.0f;
    for (int b = 0; b < nb; ++b) {                   // fixed order => deterministic
        s += psum[b * C + c]; ss += psumsq[b * C + c];
    }
    float inv  = 1.0f / (float)nRows;
    float mu   = s * inv;
    float var  = ss * inv - mu * mu;                 // biased, as torch BN forward
    float rstd = rsqrtf(var + BN_EPS);
    float sc   = gamma[c] * rstd;
    scale[c] = sc;
    shift[c] = beta[c] - mu * sc;
}

// a = gelu(a*scale + shift), exact gelu (erf), in place
template<int C>
__global__ void bn_gelu_ip(float* __restrict__ a, const float* __restrict__ sc,
                           const float* __restrict__ sh, int n) {
    int gid = blockIdx.x * blockDim.x + threadIdx.x;
    if (gid >= n * C) return;
    int c = gid % C;
    float v = a[gid] * sc[c] + sh[c];
    a[gid] = 0.5f * v * (1.0f + erff(v * 0.70710678118654752f));
}

// out += a*scale + shift  (out already holds the skip branch)
template<int C>
__global__ void bn_add_out(const float* __restrict__ a, const float* __restrict__ sc,
                           const float* __restrict__ sh, float* __restrict__ out, int n) {
    int gid = blockIdx.x * blockDim.x + threadIdx.x;
    if (gid >= n * C) return;
    int c = gid % C;
    out[gid] += a[gid] * sc[c] + sh[c];
}

// ---------------------------------------------------------------------------

extern "C" void kernel_launch(void* const* d_in, const int* in_sizes, int n_in,
                              void* d_out, int out_size, void* d_ws, size_t ws_size,
                              hipStream_t stream) {
    const float* x   = (const float*)d_in[0];
    const int*   ei  = (const int*)  d_in[1];
    const float* ew  = (const float*)d_in[2];
    const float* W1  = (const float*)d_in[3];
    const float* b1  = (const float*)d_in[4];
    const float* W2  = (const float*)d_in[5];
    const float* b2  = (const float*)d_in[6];
    const float* g1  = (const float*)d_in[7];
    const float* be1 = (const float*)d_in[8];
    const float* g2  = (const float*)d_in[9];
    const float* be2 = (const float*)d_in[10];
    const float* Wsk = (const float*)d_in[11];
    const float* bsk = (const float*)d_in[12];
    float* out = (float*)d_out;

    const int N = in_sizes[0] / IN_C;
    const int E = in_sizes[2];
    const int* rowI = ei;         // edge_index[0]
    const int* colI = ei + E;     // edge_index[1]

    // workspace layout (floats)
    float* ws    = (float*)d_ws;
    float* dinv  = ws;                               // N
    float* h1    = dinv + N;                         // N*C1 (later: h2 [0,N*C2), a2 [N*C2,N*C1))
    float* a1    = h1 + (size_t)N * C1;              // N*C1 (becomes x1 in place)
    float* psum  = a1 + (size_t)N * C1;              // NB_STATS*C1
    float* psss  = psum + (size_t)NB_STATS * C1;     // NB_STATS*C1
    float* scale = psss + (size_t)NB_STATS * C1;     // C1
    float* shift = scale + C1;                       // C1
    float* h2 = h1;
    float* a2 = h1 + (size_t)N * C2;

    const int rpb = (N + NB_STATS - 1) / NB_STATS;   // rows per stats block

    // symmetric normalization: deg -> dinv
    fill_ones <<<(N + 255) / 256, 256, 0, stream>>>(dinv, N);
    deg_accum <<<(E + 255) / 256, 256, 0, stream>>>(colI, ew, dinv, E);
    rsqrt_ip  <<<(N + 255) / 256, 256, 0, stream>>>(dinv, N);

    // conv1: h1 = x @ W1 ; a1 = aggregate(h1) + b1 ; x1 = gelu(BN1(a1))
    gemm_wmma_f32<IN_C, C1><<<(N + 63) / 64, 128, 0, stream>>>(x, W1, nullptr, h1, N);
    selfloop_init<C1><<<((N * C1) + 255) / 256, 256, 0, stream>>>(h1, dinv, b1, a1, N);
    edge_scatter<C1><<<((E * (C1 / 4)) + 255) / 256, 256, 0, stream>>>(h1, rowI, colI, ew, dinv, a1, E);
    bn_partial<C1><<<NB_STATS, 256, 0, stream>>>(a1, N, rpb, psum, psss);
    bn_finalize<C1><<<1, C1, 0, stream>>>(psum, psss, g1, be1, scale, shift, N, NB_STATS);
    bn_gelu_ip<C1><<<((N * C1) + 255) / 256, 256, 0, stream>>>(a1, scale, shift, N);

    // conv2: h2 = x1 @ W2 ; a2 = aggregate(h2) + b2
    gemm_wmma_f32<C1, C2><<<(N + 63) / 64, 128, 0, stream>>>(a1, W2, nullptr, h2, N);
    selfloop_init<C2><<<((N * C2) + 255) / 256, 256, 0, stream>>>(h2, dinv, b2, a2, N);
    edge_scatter<C2><<<((E * (C2 / 4)) + 255) / 256, 256, 0, stream>>>(h2, rowI, colI, ew, dinv, a2, E);

    // skip: out = x @ Ws + bs
    gemm_wmma_f32<IN_C, C2><<<(N + 63) / 64, 128, 0, stream>>>(x, Wsk, bsk, out, N);

    // out += BN2(a2)
    bn_partial<C2><<<NB_STATS, 256, 0, stream>>>(a2, N, rpb, psum, psss);
    bn_finalize<C2><<<1, C2, 0, stream>>>(psum, psss, g2, be2, scale, shift, N, NB_STATS);
    bn_add_out<C2><<<((N * C2) + 255) / 256, 256, 0, stream>>>(a2, scale, shift, out, N);
}